// SecondOrderTaylorRBF_22196390986304
// MI455X (gfx1250) — compile-verified
//
#include <hip/hip_runtime.h>

#define DD 8     // data dimension
#define TN 16    // n-tile (WMMA M)
#define TM 16    // m-tile per wave (WMMA N)
#define WPB 8    // waves per block (wave32)

typedef __attribute__((ext_vector_type(2))) float v2f;
typedef __attribute__((ext_vector_type(8))) float v8f;

static __device__ __forceinline__ v8f wmma_f32_k4(v2f a, v2f b, v8f c) {
  // D = A(16x4,f32) * B(4x16,f32) + C(16x16,f32)
  return __builtin_amdgcn_wmma_f32_16x16x4_f32(
      /*neg_a=*/false, a, /*neg_b=*/false, b,
      /*c_mod=*/(short)0, c, /*reuse_a=*/false, /*reuse_b=*/false);
}

__global__ __launch_bounds__(256, 1)
void taylor_rbf_kernel(const float* __restrict__ x1, const float* __restrict__ x2,
                       const float* __restrict__ Lm, const float* __restrict__ sg,
                       float* __restrict__ out, int n, int m) {
  __shared__ float s_inv2[DD][DD];            // inv2[j][k] = 1/L[j,k]^2
  __shared__ float s_M[DD][DD], s_M2[DD][DD]; // M[c][j] = inv2[0][c] + inv2[j][c]
  __shared__ float s_A[DD], s_l2[DD], s_tl[DD], s_tl2[DD], s_c1[DD], s_c0[DD];
  __shared__ float s_x1[TN][DD];              // x1 tile (shared by all 8 waves)
  __shared__ float s_a1[TN][DD];              // a1[r][j] = sum_k inv2[j][k]*x1[r][k]^2
  __shared__ float s_x2[WPB][TM][DD];         // per-wave x2 tile
  __shared__ float s_a2[WPB][TM][DD];         // per-wave a2[r][j]

  const int tid  = threadIdx.x;
  const int lane = tid & 31;
  const int wv   = tid >> 5;
  const int n0   = blockIdx.y * TN;
  const int m0   = (blockIdx.x * WPB + wv) * TM;

  // ---- phase 1: raw loads + inv2 ----
  if (tid < 64) {
    int j = tid >> 3, k = tid & 7;
    float L = Lm[j * DD + k];
    s_inv2[j][k] = 1.0f / (L * L);
  }
  if (tid >= 128) {
    int idx = tid - 128;
    int r = idx >> 3, k = idx & 7;
    s_x1[r][k] = (n0 + r < n) ? x1[(size_t)(n0 + r) * DD + k] : 0.0f;
  }
#pragma unroll
  for (int i = 0; i < 4; ++i) {
    int e = lane * 4 + i;
    int r = e >> 3, k = e & 7;
    s_x2[wv][r][k] = (m0 + r < m) ? x2[(size_t)(m0 + r) * DD + k] : 0.0f;
  }
  __syncthreads();

  // ---- phase 2: derived parameter tables + per-row quadratics ----
  if (tid < 128) {              // a1[r][j]
    int r = tid >> 3, j = tid & 7;
    float s = 0.f;
#pragma unroll
    for (int k = 0; k < DD; ++k) { float v = s_x1[r][k]; s = __builtin_fmaf(s_inv2[j][k] * v, v, s); }
    s_a1[r][j] = s;
  } else if (tid < 192) {       // M, M^2
    int idx = tid - 128;
    int c = idx >> 3, j = idx & 7;
    float Mv = s_inv2[0][c] + s_inv2[j][c];
    s_M[c][j] = Mv; s_M2[c][j] = Mv * Mv;
  } else if (tid < 200) {       // per-j scalars
    int j = tid - 192;
    float L0  = Lm[j];                      // L[I0=0][j]
    float l2  = L0 * L0;
    float tl  = s_inv2[0][j] + s_inv2[j][j];
    float tl2 = tl * tl;
    s_l2[j] = l2; s_tl[j] = tl; s_tl2[j] = tl2;
    s_A[j]  = sg[j] / (l2 * l2);
    s_c1[j] = 5.0f * tl + l2 * tl2;         // poly = tl2*d2^2 - c1*d2 + c0
    s_c0[j] = 2.0f + l2 * tl;
  }
#pragma unroll
  for (int i = 0; i < 4; ++i) { // a2[r][j], per wave
    int e = lane * 4 + i;
    int r = e >> 3, j = e & 7;
    float s = 0.f;
#pragma unroll
    for (int k = 0; k < DD; ++k) { float v = s_x2[wv][r][k]; s = __builtin_fmaf(s_inv2[j][k] * v, v, s); }
    s_a2[wv][r][j] = s;
  }
  __syncthreads();

  if (m0 >= m || n0 >= n) return;   // wave-uniform; no barriers past this point

  // ---- WMMA: cross_j[n,m] = sum_k (x1[n,k]*inv2[j,k]) * x2[m,k], 8 GEMM-K8 tiles ----
  const int lc = lane & 15;          // A row / B col / C col
  const int kb = (lane >> 4) * 2;    // K base for this half-wave (0 or 2)

  v2f blo, bhi;                      // B fragments: B[k][col] = x2[col][k]
  blo.x = s_x2[wv][lc][kb];     blo.y = s_x2[wv][lc][kb + 1];
  bhi.x = s_x2[wv][lc][4 + kb]; bhi.y = s_x2[wv][lc][4 + kb + 1];
  const float xa0 = s_x1[lc][kb],     xa1 = s_x1[lc][kb + 1];
  const float xa4 = s_x1[lc][4 + kb], xa5 = s_x1[lc][4 + kb + 1];

  v8f cross[DD];
#pragma unroll
  for (int j = 0; j < DD; ++j) {
    v2f alo, ahi;                    // A fragments: x1 row scaled by inv2[j][k]
    alo.x = xa0 * s_inv2[j][kb];     alo.y = xa1 * s_inv2[j][kb + 1];
    ahi.x = xa4 * s_inv2[j][4 + kb]; ahi.y = xa5 * s_inv2[j][4 + kb + 1];
    v8f acc = {0.f, 0.f, 0.f, 0.f, 0.f, 0.f, 0.f, 0.f};
    acc = wmma_f32_k4(alo, blo, acc);   // K = 0..3
    acc = wmma_f32_k4(ahi, bhi, acc);   // K = 4..7
    cross[j] = acc;
  }

  // ---- epilogue: per pair (C-layout: vgpr r -> row r+8*hi, lane -> col) ----
  const int col = lc;
  const int hi  = lane >> 4;
  float x2v[DD], a2v[DD];
#pragma unroll
  for (int k = 0; k < DD; ++k) x2v[k] = s_x2[wv][col][k];
#pragma unroll
  for (int j = 0; j < DD; ++j) a2v[j] = s_a2[wv][col][j];
  const float sigma0 = sg[0];

#pragma unroll
  for (int r = 0; r < 8; ++r) {
    const int row = r + 8 * hi;
    float d2[DD];
#pragma unroll
    for (int k = 0; k < DD; ++k) { float t = s_x1[row][k] - x2v[k]; d2[k] = t * t; }
    float E[DD];
#pragma unroll
    for (int j = 0; j < DD; ++j) {
      float s = s_a1[row][j] + a2v[j] - 2.0f * cross[j][r];   // s_j = ||x1-x2||^2_{inv2_j}
      E[j] = __expf(-0.5f * s);
    }
    float acc = 0.0f;
#pragma unroll
    for (int j = 0; j < DD; ++j) {
      float P = 0.0f;                                   // P_j = sum_c E_c*(M - M^2 d2[c])
#pragma unroll
      for (int c = 0; c < DD; ++c)
        P = __builtin_fmaf(E[c], __builtin_fmaf(-s_M2[c][j], d2[c], s_M[c][j]), P);
      const float Ej   = E[j];
      const float Ej2  = Ej * Ej;
      const float lmd  = s_l2[j] - d2[j];
      const float t1   = s_A[j] * Ej * lmd * P;                                    // off-diag+diag raw
      const float dOld = s_A[j] * Ej2 * (s_tl[j] - s_tl2[j] * d2[j]) * lmd;        // remove raw diag
      const float poly = __builtin_fmaf(d2[j],
                          __builtin_fmaf(s_tl2[j], d2[j], -s_c1[j]), s_c0[j]);
      const float dNew = s_A[j] * Ej2 * poly;                                      // add desired diag
      acc += t1 - dOld + dNew;
    }
    if (n0 + row < n)
      out[(size_t)(n0 + row) * m + (m0 + col)] = sigma0 * E[0] * acc;
  }
}

extern "C" void kernel_launch(void* const* d_in, const int* in_sizes, int n_in,
                              void* d_out, int out_size, void* d_ws, size_t ws_size,
                              hipStream_t stream) {
  const float* x1 = (const float*)d_in[0];
  const float* x2 = (const float*)d_in[1];
  const float* Lm = (const float*)d_in[2];   // lengthscale [8,8]
  const float* sg = (const float*)d_in[3];   // sigma [8]
  float* out = (float*)d_out;
  const int n = in_sizes[0] / DD;
  const int m = in_sizes[1] / DD;
  dim3 grid((m + TM * WPB - 1) / (TM * WPB), (n + TN - 1) / TN);
  taylor_rbf_kernel<<<grid, dim3(256), 0, stream>>>(x1, x2, Lm, sg, out, n, m);
}